// L_color_gaussian_16999480558070
// MI455X (gfx1250) — compile-verified
//
#include <hip/hip_runtime.h>
#include <math.h>

typedef __attribute__((ext_vector_type(2))) float v2f;
typedef __attribute__((ext_vector_type(8))) float v8f;
typedef __attribute__((ext_vector_type(4))) unsigned int u32x4;
typedef __attribute__((ext_vector_type(8))) int i32x8;
typedef __attribute__((ext_vector_type(4))) int i32x4;

namespace {
constexpr int kB = 16, kC = 3, kH = 512, kW = 512;
constexpr int kRad = 3;
constexpr int kHV = kH - 6, kWV = kW - 6;          // 506 valid SSIM extent
constexpr float kGs = -0.5f / (100.0f * 100.0f);   // spatial gaussian coeff
constexpr float kGc = -0.5f / (100.0f * 100.0f);   // color gaussian coeff
constexpr float kC1 = 0.0004f;                     // (0.01*2)^2
constexpr float kC2 = 0.0036f;                     // (0.03*2)^2
constexpr float kCovN = 49.0f / 48.0f;             // n/(n-1)
}

__device__ __forceinline__ int reflect101(int p, int n) {
  p = (p < 0) ? -p : p;
  p = (p >= n) ? (2 * n - 2 - p) : p;
  return p;
}

// ---------------------------------------------------------------------------
// Stage 1: bilateral filter (29-tap circular window, cross-channel L1).
// Pure fp32 LDS stencil; transcendental-bound (v_exp_f32), not WMMA material.
// ---------------------------------------------------------------------------
#define BTX 32
#define BTY 8

__global__ __launch_bounds__(256) void bilateral_kernel(
    const float* __restrict__ in, float* __restrict__ out) {
  __shared__ float tile[kC][BTY + 6][BTX + 8];
  __shared__ float swtab[10];
  const int tid = threadIdx.x;
  const int b = blockIdx.z;
  const int bx0 = blockIdx.x * BTX;
  const int by0 = blockIdx.y * BTY;
  const float* img = in + (size_t)b * kC * kH * kW;

  if (tid < 10) swtab[tid] = __expf(kGs * (float)tid);
  const int loadN = kC * (BTY + 6) * (BTX + 6);
  for (int idx = tid; idx < loadN; idx += 256) {
    int c = idx / ((BTY + 6) * (BTX + 6));
    int r = idx % ((BTY + 6) * (BTX + 6));
    int ly = r / (BTX + 6);
    int lx = r % (BTX + 6);
    int gy = reflect101(by0 - kRad + ly, kH);
    int gx = reflect101(bx0 - kRad + lx, kW);
    tile[c][ly][lx] = img[(c * kH + gy) * kW + gx];
  }
  __syncthreads();

  const int tx = tid & (BTX - 1);
  const int ty = tid >> 5;
  const int lx = tx + kRad, ly = ty + kRad;
  const float x0 = tile[0][ly][lx];
  const float x1 = tile[1][ly][lx];
  const float x2 = tile[2][ly][lx];
  float n0 = 0.f, n1 = 0.f, n2 = 0.f, den = 0.f;
#pragma unroll
  for (int di = -kRad; di <= kRad; ++di) {
#pragma unroll
    for (int dj = -kRad; dj <= kRad; ++dj) {
      constexpr int R2 = kRad * kRad;
      const int r2 = di * di + dj * dj;
      if (r2 > R2) continue;  // circular 29-tap window, matches reference
      const float p0 = tile[0][ly + di][lx + dj];
      const float p1 = tile[1][ly + di][lx + dj];
      const float p2 = tile[2][ly + di][lx + dj];
      const float d1 = fabsf(p0 - x0) + fabsf(p1 - x1) + fabsf(p2 - x2);
      const float w = swtab[r2] * __expf(kGc * d1 * d1);
      n0 = fmaf(w, p0, n0);
      n1 = fmaf(w, p1, n1);
      n2 = fmaf(w, p2, n2);
      den += w;
    }
  }
  const float inv = 1.0f / den;
  const int gy = by0 + ty, gx = bx0 + tx;
  float* o = out + (size_t)b * kC * kH * kW;
  o[(0 * kH + gy) * kW + gx] = n0 * inv;
  o[(1 * kH + gy) * kW + gx] = n1 * inv;
  o[(2 * kH + gy) * kW + gx] = n2 * inv;
}

// ---------------------------------------------------------------------------
// Stage 2: fused SSIM. Tile loads via Tensor Data Mover (TDM), box filters via
// f32 WMMA: Out(16x16) = A_band(16x24) * Q(24x24) * B_band(24x16) with 0/1
// 7-wide band matrices, K chunked by 4 for V_WMMA_F32_16X16X4_F32.
// One wave per workgroup so every TDM descriptor field is blockIdx-derived
// (compiler-provably uniform -> SGPRs).
// ---------------------------------------------------------------------------
#define STRD 36  // LDS row stride in floats: 24x32 tile + TDM pad of 4 dwords

// Single LDS object => known byte offsets from workgroup LDS base.
// Layout (floats): Se[24][36] @0, Sl @864, Sq @1728, Sv[16][36] @2592.
#define SE_OFF_B 0u
#define SL_OFF_B 3456u
#define SQ_OFF_B 6912u

__device__ __forceinline__ float bandf(int base, int k) {
  return (k >= base && k <= base + 6) ? 1.0f : 0.0f;
}

__device__ __forceinline__ v8f wmma_f32(v2f a, v2f b, v8f c) {
  return __builtin_amdgcn_wmma_f32_16x16x4_f32(
      /*neg_a=*/false, a, /*neg_b=*/false, b,
      /*c_mod=*/(short)0, c, /*reuse_a=*/false, /*reuse_b=*/false);
}

// TDM: DMA a 24-row x 32-col f32 tile (global, row stride 512) into LDS with
// a 4-dword pad after every 32 dwords (-> 36-float LDS rows). tensor_dim is
// the remaining extent from the tile start so OOB elements are written as 0.
__device__ __forceinline__ void tdm_load_tile(const float* gsrc,
                                              unsigned lds_byte_off,
                                              int rem_w, int rem_h) {
  const unsigned long long ga = (unsigned long long)(uintptr_t)gsrc;
  u32x4 g0;
  g0[0] = 1u;                                   // count=1, user descriptor
  g0[1] = lds_byte_off;                         // lds_addr (bytes)
  g0[2] = (unsigned)ga;                         // global_addr[31:0]
  g0[3] = (unsigned)((ga >> 32) & 0x01FFFFFFu)  // global_addr[56:32]
          | (2u << 30);                         // type=2 ("image")
  i32x8 g1;
  g1[0] = (2 << 16)        // data_size = 4 bytes
          | (1 << 20)      // pad_enable
          | (4 << 22)      // pad_interval code 4 = every 32 dwords
          | (3 << 25);     // pad_amount  code 3 = 4 dwords
  g1[1] = (rem_w & 0xFFFF) << 16;  // tensor_dim0[15:0] (remaining width)
  g1[2] = (rem_h & 0xFFFF) << 16;  // tensor_dim0 hi=0 | tensor_dim1[15:0]
  g1[3] = 32 << 16;                // tensor_dim1 hi=0 | tile_dim0 = 32
  g1[4] = 24;                      // tile_dim1 = 24, tile_dim2 = 0
  g1[5] = 512;                     // tensor_dim0_stride = 512 elements
  g1[6] = 0;
  g1[7] = 0;
  const i32x4 z4 = {0, 0, 0, 0};          // groups 2/3 unused (2-D tensor)
  const i32x8 z8 = {0, 0, 0, 0, 0, 0, 0, 0};
  __builtin_amdgcn_tensor_load_to_lds(g0, g1, z4, z4, z8, 0);
}

__global__ __launch_bounds__(32) void ssim_wmma_kernel(
    const float* __restrict__ Fe, const float* __restrict__ Fl,
    float* __restrict__ acc) {
  __shared__ float S[3 * 24 * STRD + 16 * STRD];
  float* Se = &S[0];
  float* Sl = &S[864];
  float* Sq = &S[1728];
  float* Sv = &S[2592];

  const int lane = threadIdx.x & 31;
  const int tileId = blockIdx.x;
  const int bc = tileId >> 10;             // (b,c) flat: 0..47
  const int rem = tileId & 1023;
  const int ty = rem >> 5, tx = rem & 31;  // 32x32 tiles cover 506x506
  const int gr = ty * 16, gc0 = tx * 16;

  const float* pe = Fe + (size_t)bc * kH * kW + gr * kW + gc0;
  const float* pl = Fl + (size_t)bc * kH * kW + gr * kW + gc0;

  // Two TDM DMAs fetch both 24x32 windows; OOB rows/cols arrive as zeros
  // (band weights are zero wherever valid outputs would touch the pad).
  tdm_load_tile(pe, SE_OFF_B, kW - gc0, kH - gr);
  tdm_load_tile(pl, SL_OFF_B, kW - gc0, kH - gr);
  __builtin_amdgcn_s_wait_tensorcnt(0);
  asm volatile("" ::: "memory");

  const int m = lane & 15;               // A-row / B-col / D-col index
  const int kA = (lane < 16) ? 0 : 2;    // K sub-index base for this half-wave
  const int mrow = (lane < 16) ? 0 : 8;  // D-row base for this half-wave

  // Hoisted 0/1 band constants (shared by pass-1 A and pass-2 B).
  float b0[6], b1[6];
#pragma unroll
  for (int kc = 0; kc < 6; ++kc) {
    b0[kc] = bandf(m, 4 * kc + kA);
    b1[kc] = bandf(m, 4 * kc + kA + 1);
  }

  v8f U[5];
  for (int q = 0; q < 5; ++q) {
    const float* src;
    if (q == 0) {
      src = Se;
    } else if (q == 1) {
      src = Sl;
    } else {
      for (int r = 0; r < 24; ++r) {
        const float e = Se[r * STRD + lane];
        const float l = Sl[r * STRD + lane];
        Sq[r * STRD + lane] = (q == 2) ? e * e : (q == 3) ? l * l : e * l;
      }
      src = Sq;
    }
    // Pass 1 (vertical 7-tap sums): V(16 x 32) = A_band x Q, two col-chunks.
    for (int cc = 0; cc < 2; ++cc) {
      const int c0 = cc * 16;
      v8f av = {};
#pragma unroll
      for (int kc = 0; kc < 6; ++kc) {
        v2f a, bv;
        a.x = b0[kc];
        a.y = b1[kc];
        bv.x = src[(4 * kc + kA) * STRD + c0 + m];
        bv.y = src[(4 * kc + kA + 1) * STRD + c0 + m];
        av = wmma_f32(a, bv, av);
      }
#pragma unroll
      for (int i = 0; i < 8; ++i) Sv[(mrow + i) * STRD + c0 + m] = av[i];
    }
    // Pass 2 (horizontal 7-tap sums): Out(16x16) = V x B_band.
    v8f ah = {};
#pragma unroll
    for (int kc = 0; kc < 6; ++kc) {
      v2f a, bv;
      a.x = Sv[m * STRD + 4 * kc + kA];
      a.y = Sv[m * STRD + 4 * kc + kA + 1];
      bv.x = b0[kc];  // band(k, n=m), identical band function
      bv.y = b1[kc];
      ah = wmma_f32(a, bv, ah);
    }
#pragma unroll
    for (int i = 0; i < 8; ++i) ah[i] *= (1.0f / 49.0f);  // box mean
    U[q] = ah;
  }

  // Elementwise SSIM on the five co-laid-out 16x16 results + reduction.
  float sum = 0.0f;
#pragma unroll
  for (int i = 0; i < 8; ++i) {
    const int oy = gr + mrow + i;
    const int ox = gc0 + m;
    const float ux = U[0][i], uy = U[1][i];
    const float uxx = U[2][i], uyy = U[3][i], uxy = U[4][i];
    const float vx = kCovN * (uxx - ux * ux);
    const float vy = kCovN * (uyy - uy * uy);
    const float vxy = kCovN * (uxy - ux * uy);
    const float numr = (2.0f * ux * uy + kC1) * (2.0f * vxy + kC2);
    const float denr = (ux * ux + uy * uy + kC1) * (vx + vy + kC2);
    const float s = numr / denr;
    if (oy < kHV && ox < kWV) sum += s;
  }
#pragma unroll
  for (int off = 16; off >= 1; off >>= 1) sum += __shfl_xor(sum, off, 32);
  if (lane == 0) atomicAdd(acc, sum);
}

// ---------------------------------------------------------------------------
// Stage 0/3: accumulator init + scalar finalize.
// sum_b (1 - mean_b(s)) == 16 - (sum of all s) / (3*506*506)
// ---------------------------------------------------------------------------
__global__ void zero_kernel(float* acc) {
  if (threadIdx.x == 0) acc[0] = 0.0f;
}

__global__ void finalize_kernel(const float* __restrict__ acc,
                                float* __restrict__ out) {
  out[0] = 16.0f - acc[0] / (3.0f * 506.0f * 506.0f);
}

extern "C" void kernel_launch(void* const* d_in, const int* in_sizes, int n_in,
                              void* d_out, int out_size, void* d_ws,
                              size_t ws_size, hipStream_t stream) {
  (void)in_sizes; (void)n_in; (void)out_size; (void)ws_size;
  const float* enhanced = (const float*)d_in[0];
  const float* label = (const float*)d_in[1];
  float* out = (float*)d_out;

  const size_t n = (size_t)kB * kC * kH * kW;  // 12.58M floats per image
  float* Fe = (float*)d_ws;
  float* Fl = Fe + n;
  float* acc = Fl + n;

  zero_kernel<<<1, 32, 0, stream>>>(acc);

  dim3 bgrid(kW / BTX, kH / BTY, kB);
  bilateral_kernel<<<bgrid, 256, 0, stream>>>(enhanced, Fe);
  bilateral_kernel<<<bgrid, 256, 0, stream>>>(label, Fl);

  const int tiles = kB * kC * 32 * 32;  // 49152 tiles, one wave each
  ssim_wmma_kernel<<<tiles, 32, 0, stream>>>(Fe, Fl, acc);

  finalize_kernel<<<1, 1, 0, stream>>>(acc, out);
}